// DenseGATConv_22557168238660
// MI455X (gfx1250) — compile-verified
//
#include <hip/hip_runtime.h>

// ---------------------------------------------------------------------------
// DenseGATConv-style relational conv for MI455X (gfx1250, wave32).
//   out = (adj * w_edge[edge_attr]) @ x @ W_rel + b_rel + x @ W_root
// B=16, N=1024, C_IN=C_OUT=64, E_DIM=4. All f32.
//
// Memory-bound op (adj + edge_attr = 128 MB @ 23.3 TB/s ~ 5.8us floor), so we
// keep full f32 precision and use V_WMMA_F32_16X16X4_F32 for all matmuls.
// The w_edge[edge_attr] gather goes through a 4-entry LDS table (ds_load_b32)
// to stay branchless — a ternary select chain was getting if-converted into
// exec-mask branches inside the hot loop.
// ---------------------------------------------------------------------------

typedef float v2f __attribute__((ext_vector_type(2)));
typedef float v8f __attribute__((ext_vector_type(8)));
typedef int   v2i __attribute__((ext_vector_type(2)));

#define B_            16
#define N_            1024
#define C_            64
#define WAVES_PER_WG  4
#define ROWS_PER_WAVE 16
#define ROWS_PER_WG   (WAVES_PER_WG * ROWS_PER_WAVE)   // 64

static __device__ __forceinline__ v8f wmma_f32_k4(v2f a, v2f b, v8f c) {
    return __builtin_amdgcn_wmma_f32_16x16x4_f32(
        /*neg_a=*/false, a, /*neg_b=*/false, b,
        /*c_mod=*/(short)0, c, /*reuse_a=*/false, /*reuse_b=*/false);
}

__global__ __launch_bounds__(128) void relconv_fused_kernel(
    const float* __restrict__ x,         // [B, N, C]
    const float* __restrict__ adj,       // [B, N, N]
    const int*   __restrict__ edge_attr, // [B, N, N]
    const float* __restrict__ W_rel,     // [C, C]
    const float* __restrict__ b_rel,     // [C]
    const float* __restrict__ W_root,    // [C, C]
    const float* __restrict__ w_edge,    // [E_DIM=4]
    float*       __restrict__ out)       // [B, N, C]
{
    // 4 KB per wave: aggregation tile re-staged from C/D layout to A layout.
    __shared__ float lds_agg[WAVES_PER_WG * ROWS_PER_WAVE * C_];
    __shared__ float swe[4];             // w_edge lookup table (branchless gather)

    const int tid  = threadIdx.x;
    const int wave = tid >> 5;
    const int lane = tid & 31;
    const int half = lane >> 4;    // 0: lanes 0-15, 1: lanes 16-31
    const int m    = lane & 15;

    if (tid < 4) swe[tid] = w_edge[tid];
    __syncthreads();

    const int b      = blockIdx.x / (N_ / ROWS_PER_WG);
    const int rowblk = blockIdx.x % (N_ / ROWS_PER_WG);
    const int row0   = rowblk * ROWS_PER_WG + wave * ROWS_PER_WAVE;
    const int row    = row0 + m;                    // this lane's A-matrix row

    const size_t adjRow = ((size_t)b * N_ + (size_t)row) * N_;
    const float* adjp   = adj       + adjRow + 2 * half;
    const int*   edgep  = edge_attr + adjRow + 2 * half;
    const float* xb     = x + (size_t)b * N_ * C_;

    // ------------------------------------------------------------------
    // Phase 1: agg[row0..row0+15, 0..63] = (adj*e) @ x   (K = 1024)
    // A frag (16x4 f32): lane(h,m): v0=A[m][k+2h], v1=A[m][k+2h+1]
    // B frag (4x16 f32): lane(h,m): v0=B[k+2h][n],  v1=B[k+2h+1][n]
    // ------------------------------------------------------------------
    v8f acc0 = {}, acc1 = {}, acc2 = {}, acc3 = {};

#pragma unroll 4
    for (int k = 0; k < N_; k += 4) {
        v2f araw = *(const v2f*)(adjp + k);
        v2i idx  = *(const v2i*)(edgep + k);
        const float e0 = swe[idx[0] & 3];   // ds_load_b32 gather, branchless
        const float e1 = swe[idx[1] & 3];
        v2f a;
        a[0] = araw[0] * e0;
        a[1] = araw[1] * e1;

        const float* xr = xb + (size_t)(k + 2 * half) * C_ + m;   // row K=k+2h, col n
        v2f bf0, bf1, bf2, bf3;
        bf0[0] = xr[0];   bf0[1] = xr[C_ + 0];
        bf1[0] = xr[16];  bf1[1] = xr[C_ + 16];
        bf2[0] = xr[32];  bf2[1] = xr[C_ + 32];
        bf3[0] = xr[48];  bf3[1] = xr[C_ + 48];

        acc0 = wmma_f32_k4(a, bf0, acc0);
        acc1 = wmma_f32_k4(a, bf1, acc1);
        acc2 = wmma_f32_k4(a, bf2, acc2);
        acc3 = wmma_f32_k4(a, bf3, acc3);
    }

    // ------------------------------------------------------------------
    // Stage aggregation tile to LDS (C/D layout -> row-major 16x64).
    // D VGPR r: lanes 0-15 -> (M=r, N=lane); lanes 16-31 -> (M=r+8, N=lane-16)
    // ------------------------------------------------------------------
    float* lag = lds_agg + wave * ROWS_PER_WAVE * C_;
#pragma unroll
    for (int r = 0; r < 8; ++r) {
        const int mr = r + 8 * half;
        lag[mr * C_ + m +  0] = acc0[r];
        lag[mr * C_ + m + 16] = acc1[r];
        lag[mr * C_ + m + 32] = acc2[r];
        lag[mr * C_ + m + 48] = acc3[r];
    }
    __syncthreads();

    // ------------------------------------------------------------------
    // Phase 2: out_tile = agg @ W_rel + b_rel + x_rows @ W_root  (K = 64)
    // Bias pre-loaded into C: bias depends only on column N = 16*t + m.
    // ------------------------------------------------------------------
    v8f o0, o1, o2, o3;
    {
        const float br0 = b_rel[m +  0];
        const float br1 = b_rel[m + 16];
        const float br2 = b_rel[m + 32];
        const float br3 = b_rel[m + 48];
#pragma unroll
        for (int r = 0; r < 8; ++r) { o0[r] = br0; o1[r] = br1; o2[r] = br2; o3[r] = br3; }
    }

    const float* xrowA = xb + (size_t)row * C_ + 2 * half;  // x as A-matrix rows

#pragma unroll 2
    for (int kk = 0; kk < C_; kk += 4) {
        const int kh = kk + 2 * half;

        v2f a1 = *(const v2f*)(lag + m * C_ + kh);   // agg fragment (LDS, 8B aligned)
        v2f a2 = *(const v2f*)(xrowA + kk);          // x fragment (global)

        const float* wr = W_rel  + (size_t)kh * C_ + m;
        const float* wt = W_root + (size_t)kh * C_ + m;

        v2f br0, br1, br2, br3, bt0, bt1, bt2, bt3;
        br0[0] = wr[0];   br0[1] = wr[C_ + 0];
        br1[0] = wr[16];  br1[1] = wr[C_ + 16];
        br2[0] = wr[32];  br2[1] = wr[C_ + 32];
        br3[0] = wr[48];  br3[1] = wr[C_ + 48];
        bt0[0] = wt[0];   bt0[1] = wt[C_ + 0];
        bt1[0] = wt[16];  bt1[1] = wt[C_ + 16];
        bt2[0] = wt[32];  bt2[1] = wt[C_ + 32];
        bt3[0] = wt[48];  bt3[1] = wt[C_ + 48];

        o0 = wmma_f32_k4(a1, br0, o0);  o0 = wmma_f32_k4(a2, bt0, o0);
        o1 = wmma_f32_k4(a1, br1, o1);  o1 = wmma_f32_k4(a2, bt1, o1);
        o2 = wmma_f32_k4(a1, br2, o2);  o2 = wmma_f32_k4(a2, bt2, o2);
        o3 = wmma_f32_k4(a1, br3, o3);  o3 = wmma_f32_k4(a2, bt3, o3);
    }

    // ------------------------------------------------------------------
    // Store output tile (C/D layout -> row-major global).
    // ------------------------------------------------------------------
    float* op = out + ((size_t)b * N_ + (size_t)row0) * C_;
#pragma unroll
    for (int r = 0; r < 8; ++r) {
        const int mr = r + 8 * half;
        op[(size_t)mr * C_ + m +  0] = o0[r];
        op[(size_t)mr * C_ + m + 16] = o1[r];
        op[(size_t)mr * C_ + m + 32] = o2[r];
        op[(size_t)mr * C_ + m + 48] = o3[r];
    }
}

extern "C" void kernel_launch(void* const* d_in, const int* in_sizes, int n_in,
                              void* d_out, int out_size, void* d_ws, size_t ws_size,
                              hipStream_t stream) {
    (void)in_sizes; (void)n_in; (void)out_size; (void)d_ws; (void)ws_size;

    const float* x         = (const float*)d_in[0];
    const float* adj       = (const float*)d_in[1];
    const int*   edge_attr = (const int*)d_in[2];
    const float* W_rel     = (const float*)d_in[3];
    const float* b_rel     = (const float*)d_in[4];
    const float* W_root    = (const float*)d_in[5];
    const float* w_edge    = (const float*)d_in[6];
    float*       out       = (float*)d_out;

    dim3 grid(B_ * (N_ / ROWS_PER_WG));   // 16 batches * 16 row-blocks = 256 WGs
    dim3 block(WAVES_PER_WG * 32);        // 4 wave32s
    relconv_fused_kernel<<<grid, block, 0, stream>>>(
        x, adj, edge_attr, W_rel, b_rel, W_root, w_edge, out);
}